// Seq2SeqLSTM_54881092108969
// MI455X (gfx1250) — compile-verified
//
#include <hip/hip_runtime.h>

// Seq2Seq LSTM on MI455X (gfx1250, wave32).
// Strategy: weights are L2-resident (16MB << 192MB L2); the recurrence is the
// critical path, so each of the 384 steps is one fused WMMA-GEMM+gates kernel.
// f32 WMMA (V_WMMA_F32_16X16X4_F32) keeps exact f32 precision through the
// 384-step recurrent chain (bf16 would drift via the cell state).

typedef __attribute__((ext_vector_type(2))) float v2f;
typedef __attribute__((ext_vector_type(8))) float v8f;

#define Bn  128   // batch
#define Hn  1024  // hidden
#define Sn  256   // source length
#define INn 512   // encoder input size
#define Tn  128   // target length

__device__ __forceinline__ v8f wmma4(v2f a, v2f b, v8f c) {
  // D = A(16x4,f32) * B(4x16,f32) + C(16x16,f32)
  return __builtin_amdgcn_wmma_f32_16x16x4_f32(
      /*neg_a=*/false, a, /*neg_b=*/false, b,
      /*c_mod=*/(short)0, c, /*reuse_a=*/false, /*reuse_b=*/false);
}

__device__ __forceinline__ float sigm(float x) {
  return 1.0f / (1.0f + __expf(-x));
}

// Accumulate acc[2][4] += A(2 m-tiles) * W^T(4 gate n-tiles) over K.
// A rows: Arow0/Arow1 point at row (m0+lm) / (m0+16+lm) of the activation,
// with leading dimension lda. W is [4H, ldw] row-major; B[k][n] = W[(g*H+j0+n)*ldw + k].
__device__ __forceinline__ void mma_gates(
    const float* __restrict__ Arow0, const float* __restrict__ Arow1,
    const float* __restrict__ W, long ldw, int K,
    int j0, int lm, int half, v8f acc[2][4]) {
  const float* a0 = Arow0 + 2 * half;           // lanes 0-15: K{0,1}; 16-31: K{2,3}
  const float* a1 = Arow1 + 2 * half;
  const float* b0 = W + (long)(0 * Hn + j0 + lm) * ldw + 2 * half;
  const float* b1 = W + (long)(1 * Hn + j0 + lm) * ldw + 2 * half;
  const float* b2 = W + (long)(2 * Hn + j0 + lm) * ldw + 2 * half;
  const float* b3 = W + (long)(3 * Hn + j0 + lm) * ldw + 2 * half;
  for (int kb = 0; kb < K; kb += 4) {
    v2f aA = *(const v2f*)(a0 + kb);
    v2f aB = *(const v2f*)(a1 + kb);
    v2f w0 = *(const v2f*)(b0 + kb);
    v2f w1 = *(const v2f*)(b1 + kb);
    v2f w2 = *(const v2f*)(b2 + kb);
    v2f w3 = *(const v2f*)(b3 + kb);
    acc[0][0] = wmma4(aA, w0, acc[0][0]);
    acc[1][0] = wmma4(aB, w0, acc[1][0]);
    acc[0][1] = wmma4(aA, w1, acc[0][1]);
    acc[1][1] = wmma4(aB, w1, acc[1][1]);
    acc[0][2] = wmma4(aA, w2, acc[0][2]);
    acc[1][2] = wmma4(aB, w2, acc[1][2]);
    acc[0][3] = wmma4(aA, w3, acc[0][3]);
    acc[1][3] = wmma4(aB, w3, acc[1][3]);
  }
}

// One LSTM step, fully fused: z = x@Wih^T + h@Whh^T + (bih+bhh); gates; c,h update.
// Grid: (H/16, B/32) blocks x 32 threads (1 wave). Each wave owns a 32(M)x16(N)
// patch across all 4 gates, so the elementwise gate math fuses in-register.
__global__ __launch_bounds__(32) void lstm_cell_wmma(
    const float* __restrict__ x, long xstride, int Kx,
    const float* __restrict__ Wih,   // [4H, Kx]
    const float* __restrict__ Whh,   // [4H, H]
    const float* __restrict__ bih, const float* __restrict__ bhh,
    const float* __restrict__ h_in,  // [B, H]
    float* __restrict__ c,           // [B, H], updated in place
    float* __restrict__ h_out)       // [B, H]
{
  const int lane = threadIdx.x & 31;
  const int half = lane >> 4;
  const int lm   = lane & 15;
  const int j0   = blockIdx.x * 16;
  const int m0   = blockIdx.y * 32;

  // init accumulators with the combined bias (broadcast per output column)
  v8f acc[2][4];
#pragma unroll
  for (int g = 0; g < 4; ++g) {
    float bsum = bih[g * Hn + j0 + lm] + bhh[g * Hn + j0 + lm];
    v8f z;
#pragma unroll
    for (int v = 0; v < 8; ++v) z[v] = bsum;
    acc[0][g] = z;
    acc[1][g] = z;
  }

  // x @ Wih^T  (skipped when x==nullptr: decoder step 0 uses x=0)
  if (x != nullptr) {
    const float* a0 = x + (long)(m0 + lm) * xstride;
    const float* a1 = x + (long)(m0 + 16 + lm) * xstride;
    mma_gates(a0, a1, Wih, (long)Kx, Kx, j0, lm, half, acc);
  }

  // h @ Whh^T
  {
    const float* a0 = h_in + (long)(m0 + lm) * Hn;
    const float* a1 = h_in + (long)(m0 + 16 + lm) * Hn;
    mma_gates(a0, a1, Whh, (long)Hn, Hn, j0, lm, half, acc);
  }

  // fused gate epilogue: i,f,g,o for element (m, j) live in the same lane/slot
#pragma unroll
  for (int mt = 0; mt < 2; ++mt) {
#pragma unroll
    for (int v = 0; v < 8; ++v) {
      int m = m0 + 16 * mt + v + 8 * half;  // C/D layout: M = vgpr + 8*(lane>=16)
      long idx = (long)m * Hn + j0 + lm;
      float iv = sigm(acc[mt][0][v]);
      float fv = sigm(acc[mt][1][v]);
      float gv = tanhf(acc[mt][2][v]);
      float ov = sigm(acc[mt][3][v]);
      float cn = fv * c[idx] + iv * gv;
      c[idx] = cn;
      h_out[idx] = ov * tanhf(cn);
    }
  }
}

// out[m, t, :] = h @ fc_W^T + fc_b, written directly into the (strided) output
// slice so the next decoder step reads it back as its input (no copy).
__global__ __launch_bounds__(32) void fc_wmma(
    const float* __restrict__ h,     // [B, H]
    const float* __restrict__ W,     // [H, H]
    const float* __restrict__ bias,  // [H]
    float* __restrict__ out, long out_row_stride) {
  const int lane = threadIdx.x & 31;
  const int half = lane >> 4;
  const int lm   = lane & 15;
  const int j0   = blockIdx.x * 16;
  const int m0   = blockIdx.y * 32;

  float bsum = bias[j0 + lm];
  v8f acc0, acc1;
#pragma unroll
  for (int v = 0; v < 8; ++v) { acc0[v] = bsum; acc1[v] = bsum; }

  const float* a0 = h + (long)(m0 + lm) * Hn + 2 * half;
  const float* a1 = h + (long)(m0 + 16 + lm) * Hn + 2 * half;
  const float* b  = W + (long)(j0 + lm) * Hn + 2 * half;
  for (int kb = 0; kb < Hn; kb += 4) {
    v2f aA = *(const v2f*)(a0 + kb);
    v2f aB = *(const v2f*)(a1 + kb);
    v2f wb = *(const v2f*)(b + kb);
    acc0 = wmma4(aA, wb, acc0);
    acc1 = wmma4(aB, wb, acc1);
  }

#pragma unroll
  for (int v = 0; v < 8; ++v) {
    int mA = m0 + v + 8 * half;
    int mB = mA + 16;
    out[(long)mA * out_row_stride + j0 + lm] = acc0[v];
    out[(long)mB * out_row_stride + j0 + lm] = acc1[v];
  }
}

__global__ void zero2_kernel(float* __restrict__ a, float* __restrict__ b, int n) {
  int i = blockIdx.x * blockDim.x + threadIdx.x;
  if (i < n) { a[i] = 0.0f; b[i] = 0.0f; }
}

extern "C" void kernel_launch(void* const* d_in, const int* in_sizes, int n_in,
                              void* d_out, int out_size, void* d_ws, size_t ws_size,
                              hipStream_t stream) {
  const float* input_seq = (const float*)d_in[0];   // [B, S, IN]
  const float* Wih_enc   = (const float*)d_in[1];   // [4H, IN]
  const float* Whh_enc   = (const float*)d_in[2];   // [4H, H]
  const float* bih_enc   = (const float*)d_in[3];
  const float* bhh_enc   = (const float*)d_in[4];
  const float* Wih_dec   = (const float*)d_in[5];   // [4H, H]
  const float* Whh_dec   = (const float*)d_in[6];   // [4H, H]
  const float* bih_dec   = (const float*)d_in[7];
  const float* bhh_dec   = (const float*)d_in[8];
  const float* fc_W      = (const float*)d_in[9];   // [H, H]
  const float* fc_b      = (const float*)d_in[10];
  // d_in[11] = target_length (known 128, Tn)

  float* out = (float*)d_out;                        // [B, Tn, H]
  float* hA  = (float*)d_ws;                         // ping
  float* hB  = hA + (size_t)Bn * Hn;                 // pong
  float* cb  = hB + (size_t)Bn * Hn;                 // cell state (in place)

  zero2_kernel<<<(Bn * Hn + 255) / 256, 256, 0, stream>>>(hA, cb, Bn * Hn);

  dim3 grid(Hn / 16, Bn / 32);  // 64 x 4 = 256 waves per step

  float* hin  = hA;
  float* hout = hB;

  // ---- encoder: 256 sequential fused cell steps ----
  for (int t = 0; t < Sn; ++t) {
    lstm_cell_wmma<<<grid, 32, 0, stream>>>(
        input_seq + (long)t * INn, (long)Sn * INn, INn,
        Wih_enc, Whh_enc, bih_enc, bhh_enc, hin, cb, hout);
    float* tmp = hin; hin = hout; hout = tmp;
  }

  // ---- decoder: 128 autoregressive steps; FC output fed back via d_out ----
  for (int t = 0; t < Tn; ++t) {
    const float* x = (t == 0) ? nullptr : (out + (long)(t - 1) * Hn);
    lstm_cell_wmma<<<grid, 32, 0, stream>>>(
        x, (long)Tn * Hn, Hn,
        Wih_dec, Whh_dec, bih_dec, bhh_dec, hin, cb, hout);
    fc_wmma<<<grid, 32, 0, stream>>>(
        hout, fc_W, fc_b, out + (long)t * Hn, (long)Tn * Hn);
    float* tmp = hin; hin = hout; hout = tmp;
  }
}